// calculate_IoU_40381282517531
// MI455X (gfx1250) — compile-verified
//
#include <hip/hip_runtime.h>

// Problem geometry (fixed by the reference).
#define B_   8
#define C_   21
#define HW_  (512 * 512)

// Tiling.
#define P_               128                 // pixels per TDM tile (per wave)
#define WAVES_PER_BLOCK  8                   // 256 threads
#define BLOCKS_PER_BATCH 32
#define TILE_F           (C_ * P_)           // 2688 floats per tile
#define TILE_BYTES       (TILE_F * 4)        // 10752 B
#define SMEM_BYTES       (2 * WAVES_PER_BLOCK * TILE_BYTES)  // 172032 B (double buffer)

static_assert(HW_ % (BLOCKS_PER_BATCH * WAVES_PER_BLOCK * P_) == 0, "tiling must divide HW");

typedef __attribute__((ext_vector_type(4))) unsigned int u32x4;
typedef __attribute__((ext_vector_type(8))) int          i32x8;
typedef __attribute__((ext_vector_type(4))) int          i32x4;

// Issue one TDM 2D-tile load: tile_dim0 = P_ contiguous floats (pixels),
// tile_dim1 = C_ rows with row stride HW_ floats, landing contiguously in LDS
// at byte offset lds_byte_off (row c at lds_byte_off + c*P_*4).
__device__ __forceinline__ void tdm_load_tile(unsigned lds_byte_off,
                                              unsigned long long gaddr_bytes) {
  u32x4 g0;
  g0.x = 1u;                                            // count=1, user-mode, gather off
  g0.y = lds_byte_off;                                  // lds_addr (bytes)
  g0.z = (unsigned)(gaddr_bytes & 0xFFFFFFFFull);       // global_addr[31:0]  (bits 95:64)
  g0.w = (unsigned)((gaddr_bytes >> 32) & 0x1FFFFFFull) // global_addr[56:32] (bits 120:96)
         | (2u << 30);                                  // type = 2 ("image") in bits 127:126

  i32x8 g1;
  g1[0] = (int)(2u << 16);                              // wg_mask=0, data_size=2 (4 B)
  g1[1] = (int)((unsigned)(P_ & 0xFFFF) << 16);         // tensor_dim0[15:0] in bits 63:48
  g1[2] = (int)(((P_ >> 16) & 0xFFFF)                   // tensor_dim0[31:16]
         | ((unsigned)(C_ & 0xFFFF) << 16));            // tensor_dim1[15:0]
  g1[3] = (int)(((C_ >> 16) & 0xFFFF)                   // tensor_dim1[31:16]
         | ((unsigned)(P_ & 0xFFFF) << 16));            // tile_dim0 = P_
  g1[4] = (int)(C_ & 0xFFFF);                           // tile_dim1 = C_, tile_dim2 = 0
  g1[5] = (int)(HW_);                                   // tensor_dim0_stride[31:0] (elements)
  g1[6] = 0;                                            // stride[47:32]=0, dim1_stride lo=0
  g1[7] = 0;                                            // dim1_stride hi = 0

  i32x4 z4 = {0, 0, 0, 0};                              // groups 2/3: unused (2D tile)
  i32x8 z8 = {0, 0, 0, 0, 0, 0, 0, 0};
  // amdgpu-toolchain (clang-23) 6-arg form: (g0, g1, g2, g3, g_extra, cpol)
  __builtin_amdgcn_tensor_load_to_lds(g0, g1, z4, z4, z8, 0);
}

__global__ void iou_init_kernel(unsigned* __restrict__ cnt) {
  if (threadIdx.x < 2 * B_) cnt[threadIdx.x] = 0u;
}

__global__ __launch_bounds__(WAVES_PER_BLOCK * 32)
void iou_main_kernel(const float* __restrict__ pred,
                     const int*   __restrict__ gt,
                     unsigned*    __restrict__ cnt) {
  extern __shared__ float smem[];  // [2][WAVES_PER_BLOCK][TILE_F]

  const int batch = (int)blockIdx.x / BLOCKS_PER_BATCH;
  const int blk   = (int)blockIdx.x % BLOCKS_PER_BATCH;
  const int wave  = (int)threadIdx.x >> 5;
  const int lane  = (int)threadIdx.x & 31;

  const int pixels_per_block = HW_ / BLOCKS_PER_BATCH;                 // 8192
  const int pixels_per_wave  = pixels_per_block / WAVES_PER_BLOCK;     // 1024
  const int tiles            = pixels_per_wave / P_;                   // 8
  const int base_pix         = blk * pixels_per_block + wave * pixels_per_wave;

  // LDS byte offsets of this wave's two buffers (dynamic LDS starts at 0: no static LDS).
  const unsigned slot0 = (unsigned)((0 * WAVES_PER_BLOCK + wave) * TILE_BYTES);
  const unsigned slot1 = (unsigned)((1 * WAVES_PER_BLOCK + wave) * TILE_BYTES);

  const unsigned long long pred_base =
      (unsigned long long)(uintptr_t)pred +
      4ull * ((unsigned long long)batch * C_ * HW_ + (unsigned long long)base_pix);

  // Prime the pipeline: tile 0 -> slot0.
  tdm_load_tile(slot0, pred_base);

  int tp = 0, fpfn = 0;

  for (int t = 0; t < tiles; ++t) {
    const int buf = t & 1;
    if (t + 1 < tiles) {
      // Prefetch next tile into the other buffer, then wait for tile t only.
      tdm_load_tile(buf ? slot0 : slot1,
                    pred_base + 4ull * (unsigned long long)(t + 1) * P_);
      __builtin_amdgcn_s_wait_tensorcnt(1);
    } else {
      __builtin_amdgcn_s_wait_tensorcnt(0);
    }

    const float4* tilev =
        (const float4*)(smem + (buf * WAVES_PER_BLOCK + wave) * TILE_F);

    // Each lane owns 4 consecutive pixels of this 128-pixel tile.
    float4 best = tilev[0 * (P_ / 4) + lane];   // class 0
    int bx = 0, by = 0, bz = 0, bw = 0;
#pragma unroll
    for (int c = 1; c < C_; ++c) {
      const float4 v = tilev[c * (P_ / 4) + lane];  // ds_load_b128
      if (v.x > best.x) { best.x = v.x; bx = c; }
      if (v.y > best.y) { best.y = v.y; by = c; }
      if (v.z > best.z) { best.z = v.z; bz = c; }
      if (v.w > best.w) { best.w = v.w; bw = c; }
    }

    const int gbase = batch * HW_ + base_pix + t * P_;
    const int4 g4 = ((const int4*)(gt + gbase))[lane];  // coalesced global_load_b128
    if (g4.x != 0) { if (g4.x == bx) tp++; else fpfn++; }
    if (g4.y != 0) { if (g4.y == by) tp++; else fpfn++; }
    if (g4.z != 0) { if (g4.z == bz) tp++; else fpfn++; }
    if (g4.w != 0) { if (g4.w == bw) tp++; else fpfn++; }
  }

  // Wave32 reduction, then one atomic pair per wave.
#pragma unroll
  for (int off = 16; off > 0; off >>= 1) {
    tp   += __shfl_xor(tp,   off, 32);
    fpfn += __shfl_xor(fpfn, off, 32);
  }
  if (lane == 0) {
    atomicAdd(&cnt[batch],      (unsigned)tp);
    atomicAdd(&cnt[B_ + batch], (unsigned)fpfn);
  }
}

__global__ void iou_finalize_kernel(const unsigned* __restrict__ cnt,
                                    float* __restrict__ out) {
  if (threadIdx.x == 0 && blockIdx.x == 0) {
    float acc = 0.0f;
#pragma unroll
    for (int b = 0; b < B_; ++b) {
      const float tp    = (float)cnt[b];
      const float fpfn  = (float)cnt[B_ + b];
      const float denom = fmaxf(tp + 2.0f * fpfn, 1e-8f);
      acc += tp / denom;
    }
    out[0] = acc / (float)B_;
  }
}

extern "C" void kernel_launch(void* const* d_in, const int* in_sizes, int n_in,
                              void* d_out, int out_size, void* d_ws, size_t ws_size,
                              hipStream_t stream) {
  // setup_inputs order: p (unused), gt [8,512,512] int, pred [8,21,512,512] f32.
  const int*   gt   = (const int*)d_in[1];
  const float* pred = (const float*)d_in[2];
  unsigned*    cnt  = (unsigned*)d_ws;   // 16 counters: tp[8], fpfn[8]
  float*       out  = (float*)d_out;

  iou_init_kernel<<<1, 32, 0, stream>>>(cnt);
  iou_main_kernel<<<B_ * BLOCKS_PER_BATCH, WAVES_PER_BLOCK * 32, SMEM_BYTES, stream>>>(
      pred, gt, cnt);
  iou_finalize_kernel<<<1, 32, 0, stream>>>(cnt, out);
}